// GeneralConvNet_22935125360681
// MI455X (gfx1250) — compile-verified
//
#include <hip/hip_runtime.h>
#include <hip/hip_bf16.h>

// ---------------------------------------------------------------------------
// GNN (GeneralConv x4) for MI455X / gfx1250.
//
// Math rewrite: segment_sum(x[src] @ W, dst) == segment_sum(x[src], dst) @ W,
// and the head-mean folds into the weight:  W_eff[i,c] = mean_h W[i, h*C+c].
// => per layer: LDS scatter-add over 1000 edges, then a 112x192 @ 192x192 f32
// WMMA GEMM per graph.  One graph per workgroup, everything in LDS (261 KB).
// Weights are pre-transposed to [N x K] so each lane's B fragment is one
// contiguous global_load_b64; A fragments come from LDS via ds_load_2addr.
// GEMM variants are template-specialized so every v_wmma executes with EXEC
// all-1s (ISA 7.12 requirement) and no divergence in the K-loop.
// ---------------------------------------------------------------------------

typedef __attribute__((ext_vector_type(2))) float v2f;
typedef __attribute__((ext_vector_type(8))) float v8f;

#define N_GRAPHS 256
#define NODES_G  100
#define EDGES_G  1000
#define E_TOT    (N_GRAPHS * EDGES_G)
#define MPAD     112              // 100 padded to 7 x 16
#define HID      192
#define OUTC     64
#define EMB      16
#define BUF      (MPAD * HID)     // 21504 floats per LDS buffer

// LDS float offsets
#define OFF_B0   0
#define OFF_B1   BUF
#define OFF_B2   (2 * BUF)
#define OFF_SM   (3 * BUF)        // pooled[64], demo[5], h at +80
#define OFF_EDG  (OFF_SM + 256)   // 2000 ints (local src, dst)
#define SMEM_FLOATS (OFF_EDG + 2000)

// d_ws float offsets: transposed head-averaged weights [N x K] + biases
#define WS_WT0 0                          // [192 x 16]
#define WS_WT1 (WS_WT0 + HID * EMB)       // [192 x 192]
#define WS_WT2 (WS_WT1 + HID * HID)       // [192 x 192]
#define WS_WT3 (WS_WT2 + HID * HID)       // [64  x 192]
#define WS_ST0 (WS_WT3 + OUTC * HID)      // w_self0^T [192 x 16]
#define WS_ST3 (WS_ST0 + HID * EMB)       // w_self3^T [64 x 192]
#define WS_B0  (WS_ST3 + OUTC * HID)
#define WS_B1  (WS_B0 + HID)
#define WS_B2  (WS_B1 + HID)
#define WS_B3  (WS_B2 + HID)

// -------- prep kernels: fold head-mean into weights, store transposed ------

// wt[c*fanin + r] = mean_h wm[r, h*C + c]   (wm is [fanin, 4*C])
__global__ void prep_wt(const float* __restrict__ wm, float* __restrict__ wt,
                        int fanin, int C) {
  int i = blockIdx.x * blockDim.x + threadIdx.x;
  if (i >= fanin * C) return;
  int c = i / fanin, r = i % fanin;
  const float* p = wm + (size_t)r * 4 * C + c;
  wt[i] = 0.25f * (p[0] + p[C] + p[2 * C] + p[3 * C]);
}

// plain transpose: wt[c*fanin + r] = w[r*C + c]   (w is [fanin, C])
__global__ void prep_t(const float* __restrict__ w, float* __restrict__ wt,
                       int fanin, int C) {
  int i = blockIdx.x * blockDim.x + threadIdx.x;
  if (i >= fanin * C) return;
  int c = i / fanin, r = i % fanin;
  wt[i] = w[(size_t)r * C + c];
}

__global__ void prep_b(const float* __restrict__ bm, const float* __restrict__ bself,
                       float* __restrict__ beff, int C) {
  int c = blockIdx.x * blockDim.x + threadIdx.x;
  if (c >= C) return;
  float v = 0.25f * (bm[c] + bm[C + c] + bm[2 * C + c] + bm[3 * C + c]);
  if (bself) v += bself[c];
  beff[c] = v;
}

// ---------------- WMMA block GEMM: D = A@B (+ A2@B2) + bias (+ add) --------
// A, A2, add, D in LDS; Bt, B2t are TRANSPOSED [N x K] in global/L2.
// M fixed = 112 (7 tiles).  Fragment layout per CDNA5 ISA
// (A 16x4, B 4x16, C/D 16x16, f32): lanes 0-15 carry K={0,1} / rows M=j,
// lanes 16-31 carry K={2,3} / rows M=j+8.

template <bool HAS_A2, bool HAS_ADD, int K, int N>
__device__ __forceinline__ void block_gemm(
    const float* __restrict__ Alds, int lda,
    const float* __restrict__ Bt,
    const float* __restrict__ A2lds, int lda2,
    const float* __restrict__ B2t,
    const float* __restrict__ biasg,
    const float* __restrict__ addLds, int ldadd,
    float* __restrict__ Dlds, int ldd,
    int wave, int lane) {
  constexpr int nT = N >> 4;
  const int l16 = lane & 15;
  const int khalf = (lane >> 4) << 1;          // 0 for lanes 0-15, 2 for 16-31
  for (int t = wave; t < 7 * nT; t += 8) {
    const int tm = t / nT;
    const int tn = t - tm * nT;
    const int n = (tn << 4) + l16;             // output column
    const int arow = (tm << 4) + l16;          // A row (M)
    const float bv = biasg[n];
    v8f c = {bv, bv, bv, bv, bv, bv, bv, bv};
    const float* aP = Alds + arow * lda + khalf;
    const float* bP = Bt + n * K + khalf;      // contiguous pair -> b64 load
    const float* a2P = HAS_A2 ? (A2lds + arow * lda2 + khalf) : nullptr;
    const float* b2P = HAS_A2 ? (B2t + n * K + khalf) : nullptr;
#pragma unroll 4
    for (int k = 0; k < K; k += 4) {
      v2f a; a.x = aP[k]; a.y = aP[k + 1];
      v2f b; b.x = bP[k]; b.y = bP[k + 1];
      c = __builtin_amdgcn_wmma_f32_16x16x4_f32(false, a, false, b,
                                                (short)0, c, false, false);
      if constexpr (HAS_A2) {
        v2f a2; a2.x = a2P[k]; a2.y = a2P[k + 1];
        v2f b2; b2.x = b2P[k]; b2.y = b2P[k + 1];
        c = __builtin_amdgcn_wmma_f32_16x16x4_f32(false, a2, false, b2,
                                                  (short)0, c, false, false);
      }
    }
    const int rbase = (tm << 4) + ((lane >> 4) << 3);   // +8 for lanes 16-31
#pragma unroll
    for (int j = 0; j < 8; ++j) {
      float v = c[j];
      const int r = rbase + j;
      if constexpr (HAS_ADD) v += addLds[r * ldadd + n];
      Dlds[r * ldd + n] = v;
    }
  }
}

// ---------------- edge scatter-add (LDS atomics) ---------------------------

__device__ __forceinline__ void scatter_add(const int* __restrict__ eS,
                                            const int* __restrict__ eD,
                                            const float* __restrict__ xin,
                                            float* __restrict__ acc,
                                            int C, int wave, int lane) {
  for (int e = wave; e < EDGES_G; e += 8) {
    const int s = eS[e], d = eD[e];
    for (int c = lane; c < C; c += 32)
      atomicAdd(&acc[d * C + c], xin[s * C + c]);
  }
}

// ---------------- main per-graph kernel ------------------------------------

__global__ __launch_bounds__(256)
void gnn_graph_kernel(const int* __restrict__ x_ids,
                      const int* __restrict__ edge_index,
                      const float* __restrict__ demographics,
                      const float* __restrict__ emb_table,
                      const float* __restrict__ w_c1, const float* __restrict__ b_c1,
                      const float* __restrict__ w_c2, const float* __restrict__ b_c2,
                      const float* __restrict__ ws, float* __restrict__ out) {
  extern __shared__ float smem[];
  float* buf0 = smem + OFF_B0;
  float* buf1 = smem + OFF_B1;
  float* buf2 = smem + OFF_B2;
  float* small = smem + OFF_SM;
  int* eS = (int*)(smem + OFF_EDG);
  int* eD = eS + EDGES_G;

  const int g = blockIdx.x;
  const int tid = threadIdx.x;
  const int wave = tid >> 5, lane = tid & 31;
  const int nbase = g * NODES_G;
  const int* srcE = edge_index + g * EDGES_G;
  const int* dstE = edge_index + E_TOT + g * EDGES_G;

  const float* wt0 = ws + WS_WT0;
  const float* wt1 = ws + WS_WT1;
  const float* wt2 = ws + WS_WT2;
  const float* wt3 = ws + WS_WT3;
  const float* st0 = ws + WS_ST0;
  const float* st3 = ws + WS_ST3;
  const float* beff0 = ws + WS_B0;
  const float* beff1 = ws + WS_B1;
  const float* beff2 = ws + WS_B2;
  const float* beff3 = ws + WS_B3;

  // warm L2 with the shared weight matrices (global_prefetch_b8)
  __builtin_prefetch(wt1 + tid * 128, 0, 1);
  __builtin_prefetch(wt2 + tid * 128, 0, 1);

  // edge list -> LDS (graph-local ids)
  for (int e = tid; e < EDGES_G; e += 256) {
    eS[e] = srcE[e] - nbase;
    eD[e] = dstE[e] - nbase;
  }
  // embedding gather -> buf0 [112x16]; zero acc -> buf1 [112x16]
  for (int i = tid; i < MPAD * EMB; i += 256) {
    const int r = i >> 4, cc = i & 15;
    buf0[i] = (r < NODES_G) ? emb_table[x_ids[nbase + r] * EMB + cc] : 0.0f;
    buf1[i] = 0.0f;
  }
  __syncthreads();

  // ---- layer 0: agg16 -> buf1; x1 = agg@W0 + x0@Wself0 + b -> buf2 ----
  scatter_add(eS, eD, buf0, buf1, EMB, wave, lane);
  __syncthreads();
  block_gemm<true, false, EMB, HID>(buf1, EMB, wt0, buf0, EMB, st0,
                                    beff0, nullptr, 0, buf2, HID, wave, lane);
  __syncthreads();

  // ---- layer 1: x2 = scatter(x1)@W1 + b + x1 -> buf0 ----
  for (int i = tid; i < MPAD * HID; i += 256) buf1[i] = 0.0f;
  __syncthreads();
  scatter_add(eS, eD, buf2, buf1, HID, wave, lane);
  __syncthreads();
  block_gemm<false, true, HID, HID>(buf1, HID, wt1, nullptr, 0, nullptr,
                                    beff1, buf2, HID, buf0, HID, wave, lane);
  __syncthreads();

  // ---- layer 2: x3 = scatter(x2)@W2 + b + x2 -> buf2 ----
  for (int i = tid; i < MPAD * HID; i += 256) buf1[i] = 0.0f;
  __syncthreads();
  scatter_add(eS, eD, buf0, buf1, HID, wave, lane);
  __syncthreads();
  block_gemm<false, true, HID, HID>(buf1, HID, wt2, nullptr, 0, nullptr,
                                    beff2, buf0, HID, buf2, HID, wave, lane);
  __syncthreads();

  // ---- layer 3: y = scatter(x3)@W3 + x3@Wself3 + b -> buf0 [112x64] ----
  for (int i = tid; i < MPAD * HID; i += 256) buf1[i] = 0.0f;
  __syncthreads();
  scatter_add(eS, eD, buf2, buf1, HID, wave, lane);
  __syncthreads();
  block_gemm<true, false, HID, OUTC>(buf1, HID, wt3, buf2, HID, st3,
                                     beff3, nullptr, 0, buf0, OUTC, wave, lane);
  __syncthreads();

  // ---- mean pool over the 100 real rows + classifier head ----
  if (tid < OUTC) {
    float s = 0.0f;
    for (int r = 0; r < NODES_G; ++r) s += buf0[r * OUTC + tid];
    small[tid] = s * (1.0f / NODES_G);
  }
  if (tid >= OUTC && tid < OUTC + 5)
    small[tid] = demographics[g * 5 + (tid - OUTC)];
  __syncthreads();
  float* hb = small + 80;
  if (tid < 64) {
    float a = b_c1[tid];
    for (int i = 0; i < OUTC + 5; ++i) a += small[i] * w_c1[i * 64 + tid];
    hb[tid] = fmaxf(a, 0.0f);
  }
  __syncthreads();
  if (tid < 2) {
    float a = b_c2[tid];
    for (int j = 0; j < 64; ++j) a += hb[j] * w_c2[j * 2 + tid];
    out[g * 2 + tid] = a;
  }
}

// ---------------- launch ----------------------------------------------------

extern "C" void kernel_launch(void* const* d_in, const int* in_sizes, int n_in,
                              void* d_out, int out_size, void* d_ws, size_t ws_size,
                              hipStream_t stream) {
  const int*   x_ids      = (const int*)d_in[0];
  const int*   edge_index = (const int*)d_in[1];
  // d_in[2] = batch (implied by graph layout, unused)
  const float* demo    = (const float*)d_in[3];
  const float* emb     = (const float*)d_in[4];
  const float* w_msg0  = (const float*)d_in[5];
  const float* b_msg0  = (const float*)d_in[6];
  const float* w_self0 = (const float*)d_in[7];
  const float* b_self0 = (const float*)d_in[8];
  const float* w_msg1  = (const float*)d_in[9];
  const float* b_msg1  = (const float*)d_in[10];
  const float* w_msg2  = (const float*)d_in[11];
  const float* b_msg2  = (const float*)d_in[12];
  const float* w_msg3  = (const float*)d_in[13];
  const float* b_msg3  = (const float*)d_in[14];
  const float* w_self3 = (const float*)d_in[15];
  const float* b_self3 = (const float*)d_in[16];
  const float* w_c1    = (const float*)d_in[17];
  const float* b_c1    = (const float*)d_in[18];
  const float* w_c2    = (const float*)d_in[19];
  const float* b_c2    = (const float*)d_in[20];
  float* ws  = (float*)d_ws;
  float* out = (float*)d_out;

  // fold head-mean into weights (transposed to [N x K]); combine biases
  prep_wt<<<(EMB * HID + 255) / 256, 256, 0, stream>>>(w_msg0, ws + WS_WT0, EMB, HID);
  prep_wt<<<(HID * HID + 255) / 256, 256, 0, stream>>>(w_msg1, ws + WS_WT1, HID, HID);
  prep_wt<<<(HID * HID + 255) / 256, 256, 0, stream>>>(w_msg2, ws + WS_WT2, HID, HID);
  prep_wt<<<(HID * OUTC + 255) / 256, 256, 0, stream>>>(w_msg3, ws + WS_WT3, HID, OUTC);
  prep_t<<<(EMB * HID + 255) / 256, 256, 0, stream>>>(w_self0, ws + WS_ST0, EMB, HID);
  prep_t<<<(HID * OUTC + 255) / 256, 256, 0, stream>>>(w_self3, ws + WS_ST3, HID, OUTC);
  prep_b<<<1, HID, 0, stream>>>(b_msg0, b_self0, ws + WS_B0, HID);
  prep_b<<<1, HID, 0, stream>>>(b_msg1, nullptr, ws + WS_B1, HID);
  prep_b<<<1, HID, 0, stream>>>(b_msg2, nullptr, ws + WS_B2, HID);
  prep_b<<<1, OUTC, 0, stream>>>(b_msg3, b_self3, ws + WS_B3, OUTC);

  const size_t smem_bytes = (size_t)SMEM_FLOATS * sizeof(float);  // ~261 KB < 320 KB/WGP
  gnn_graph_kernel<<<N_GRAPHS, 256, smem_bytes, stream>>>(
      x_ids, edge_index, demo, emb, w_c1, b_c1, w_c2, b_c2, ws, out);
}